// SelfAttention_39737037422719
// MI455X (gfx1250) — compile-verified
//
#include <hip/hip_runtime.h>
#include <hip/hip_bf16.h>

// Problem constants (match reference): B=128, N=1024, E=64, A=256
#define PB 128
#define PN 1024
#define PE 64
#define PA 256

typedef __attribute__((ext_vector_type(2))) float v2f;
typedef __attribute__((ext_vector_type(8))) float v8f;

// Workspace layout (in floats)
#define WS_QW   0
#define WS_QB   256
#define WS_KW   512
#define WS_KB   768
#define WS_VW   1024
#define WS_VB   1280
#define WS_C1   1536
#define WS_C3   1537
#define WS_PART 2048   // 512 block partials

// ---------------------------------------------------------------------------
// Kernel 1: fold weights with exact-fp32 WMMA.
//   U (16x64) = [W_emb ; b_emb ; 0...],  R = U @ W  (rows 0,1 of 16x256)
// for W in {Wq, Wk, Wv}. One wave per weight matrix, V_WMMA_F32_16X16X4_F32,
// 16 n-tiles x 16 k-steps. A fragments staged via LDS (branch-free) and
// hoisted into registers (invariant across n-tiles). Then c1=qw.kw, c3=qb.kw.
// ---------------------------------------------------------------------------
__global__ __launch_bounds__(96) void sa_prep(const float* __restrict__ W_emb,
                                              const float* __restrict__ b_emb,
                                              const float* __restrict__ Wq,
                                              const float* __restrict__ Wk,
                                              const float* __restrict__ Wv,
                                              float* __restrict__ ws) {
  // [0,1536): qw qb kw kb vw vb ; [1536,2560): staged A matrix (16x64 row-major)
  __shared__ float sh[2560];
  float* Ast = &sh[1536];

  const int tid  = threadIdx.x;
  const int wave = tid >> 5;
  const int lane = tid & 31;
  const int l0   = lane & 15;
  const int hi   = lane >> 4;

  // Stage A: row 0 = W_emb, row 1 = b_emb, rows 2..15 = 0
  for (int i = tid; i < 16 * PE; i += 96) Ast[i] = 0.0f;
  __syncthreads();
  if (tid < PE) {
    Ast[tid]      = W_emb[tid];   // row 0
    Ast[PE + tid] = b_emb[tid];   // row 1
  }
  __syncthreads();

  // Hoist A fragments: lane (l0,hi) needs A[l0][kc*4 + 2*hi + j], j=0,1
  v2f afrag[16];
#pragma unroll
  for (int kc = 0; kc < 16; ++kc) {
    const int kb = kc * 4 + 2 * hi;
    afrag[kc].x = Ast[l0 * PE + kb];
    afrag[kc].y = Ast[l0 * PE + kb + 1];
  }

  const float* __restrict__ W = (wave == 0) ? Wq : ((wave == 1) ? Wk : Wv);
  float* outw = &sh[wave * 512];        // row 0 of result (the "w" vector)
  float* outb = &sh[wave * 512 + 256];  // row 1 of result (the "b" vector)

  for (int nt = 0; nt < 16; ++nt) {
    const int nbase = nt * 16;
    v8f acc = {};
#pragma unroll
    for (int kc = 0; kc < 16; ++kc) {
      const int kb = kc * 4 + 2 * hi;
      // B (4x16 fp32): v_j = B[2*hi+j][l0] = W[kb+j][nbase+l0]
      v2f b;
      b.x = W[(kb    ) * PA + nbase + l0];
      b.y = W[(kb + 1) * PA + nbase + l0];
      acc = __builtin_amdgcn_wmma_f32_16x16x4_f32(
          /*neg_a=*/false, afrag[kc], /*neg_b=*/false, b,
          /*c_mod=*/(short)0, acc, /*reuse_a=*/false, /*reuse_b=*/false);
    }
    if (hi == 0) {                  // lanes 0-15 hold rows M=0..7 in acc[0..7]
      outw[nbase + l0] = acc[0];    // M = 0
      outb[nbase + l0] = acc[1];    // M = 1
    }
  }
  __syncthreads();

  // Copy folded vectors to workspace
  for (int i = tid; i < 1536; i += 96) ws[i] = sh[i];

  // c1 = qw.kw, c3 = qb.kw  (wave 0 partials -> LDS (reusing Ast) -> thread 0)
  if (wave == 0) {
    float p1 = 0.0f, p3 = 0.0f;
    for (int i = lane; i < PA; i += 32) {
      const float kwv = sh[512 + i];
      p1 = fmaf(sh[i],       kwv, p1);   // qw . kw
      p3 = fmaf(sh[256 + i], kwv, p3);   // qb . kw
    }
    Ast[lane]      = p1;
    Ast[64 + lane] = p3;
  }
  __syncthreads();
  if (tid == 0) {
    float c1 = 0.0f, c3 = 0.0f;
    for (int i = 0; i < 32; ++i) { c1 += Ast[i]; c3 += Ast[64 + i]; }
    ws[WS_C1] = c1;
    ws[WS_C3] = c3;
  }
}

// ---------------------------------------------------------------------------
// Kernel 2: per (b,n): alpha = (c1*x_n + c3)/16 ; s(b,n) = softmax_m(alpha*x_m) . x
// grid = B * (N/256) blocks, 256 threads; block handles 256 rows of one batch.
// Writes block-local sum of s to partial[blockIdx.x] (deterministic, no atomics).
// ---------------------------------------------------------------------------
__global__ __launch_bounds__(256) void sa_attn(const float* __restrict__ x,
                                               const float* __restrict__ wsc,
                                               float* __restrict__ partial) {
  __shared__ __align__(16) float xs[PN];
  __shared__ float red[256];
  __shared__ float red2[256];
  const int tid  = threadIdx.x;
  const int b    = blockIdx.x >> 2;   // 4 tiles per batch
  const int tile = blockIdx.x & 3;
  const float* __restrict__ xrow = x + b * PN;

  for (int i = tid; i < PN; i += 256) xs[i] = xrow[i];
  __syncthreads();

  // per-batch max / min (for a safe softmax shift)
  float mx = -3.402823466e38f, mn = 3.402823466e38f;
  for (int i = tid; i < PN; i += 256) {
    const float v = xs[i];
    mx = fmaxf(mx, v);
    mn = fminf(mn, v);
  }
  red[tid] = mx; red2[tid] = mn;
  __syncthreads();
  for (int t = 128; t > 0; t >>= 1) {
    if (tid < t) {
      red[tid]  = fmaxf(red[tid],  red[tid + t]);
      red2[tid] = fminf(red2[tid], red2[tid + t]);
    }
    __syncthreads();
  }
  const float xmax = red[0];
  const float xmin = red2[0];
  __syncthreads();  // red[] gets reused below

  const float c1 = wsc[WS_C1];
  const float c3 = wsc[WS_C3];
  const float L2E = 1.4426950408889634f;

  const float xn    = xs[tile * 256 + tid];
  const float alpha = (c1 * xn + c3) * 0.0625f;            // /sqrt(A), A=256
  const float al    = alpha * L2E;                          // exp2 domain
  const float shift = ((alpha >= 0.0f) ? alpha * xmax : alpha * xmin) * L2E;

  float s1 = 0.0f, s2 = 0.0f;
  for (int m = 0; m < PN; m += 4) {
    const float4 xm = *(const float4*)&xs[m];   // ds_load_b128 broadcast
    const float e0 = __builtin_exp2f(fmaf(al, xm.x, -shift));
    const float e1 = __builtin_exp2f(fmaf(al, xm.y, -shift));
    const float e2 = __builtin_exp2f(fmaf(al, xm.z, -shift));
    const float e3 = __builtin_exp2f(fmaf(al, xm.w, -shift));
    s1 += (e0 + e1) + (e2 + e3);
    s2 = fmaf(e0, xm.x, s2);
    s2 = fmaf(e1, xm.y, s2);
    s2 = fmaf(e2, xm.z, s2);
    s2 = fmaf(e3, xm.w, s2);
  }
  const float s = s2 / s1;

  // block sum of s over 256 rows
  red[tid] = s;
  __syncthreads();
  for (int t = 128; t > 0; t >>= 1) {
    if (tid < t) red[tid] += red[tid + t];
    __syncthreads();
  }
  if (tid == 0) partial[blockIdx.x] = red[0];
}

// ---------------------------------------------------------------------------
// Kernel 3: out[b,a] = (sum of 4 partials / N) * vw[a] + vb[a]
// grid = B blocks, 256 threads (one per a).
// ---------------------------------------------------------------------------
__global__ __launch_bounds__(256) void sa_finish(const float* __restrict__ wsc,
                                                 const float* __restrict__ partial,
                                                 float* __restrict__ out) {
  __shared__ float tsh;
  const int b   = blockIdx.x;
  const int tid = threadIdx.x;
  if (tid == 0) {
    const float t = partial[b * 4] + partial[b * 4 + 1] +
                    partial[b * 4 + 2] + partial[b * 4 + 3];
    tsh = t * (1.0f / (float)PN);
  }
  __syncthreads();
  const float t = tsh;
  out[b * PA + tid] = fmaf(t, wsc[WS_VW + tid], wsc[WS_VB + tid]);
}

// ---------------------------------------------------------------------------
extern "C" void kernel_launch(void* const* d_in, const int* in_sizes, int n_in,
                              void* d_out, int out_size, void* d_ws, size_t ws_size,
                              hipStream_t stream) {
  (void)in_sizes; (void)n_in; (void)out_size; (void)ws_size;
  const float* x     = (const float*)d_in[0];
  const float* W_emb = (const float*)d_in[1];
  const float* b_emb = (const float*)d_in[2];
  const float* Wq    = (const float*)d_in[3];
  const float* Wk    = (const float*)d_in[4];
  const float* Wv    = (const float*)d_in[5];
  float* ws  = (float*)d_ws;
  float* out = (float*)d_out;

  sa_prep<<<1, 96, 0, stream>>>(W_emb, b_emb, Wq, Wk, Wv, ws);
  sa_attn<<<(PB * PN) / 256, 256, 0, stream>>>(x, ws, ws + WS_PART);
  sa_finish<<<PB, 256, 0, stream>>>(ws, ws + WS_PART, out);
}